// SimpleGNNLayer_71683004170809
// MI455X (gfx1250) — compile-verified
//
#include <hip/hip_runtime.h>
#include <hip/hip_bf16.h>

typedef __attribute__((ext_vector_type(2))) float v2f;
typedef __attribute__((ext_vector_type(8))) float v8f;

#define DD 128          // D_IN == D_OUT == 128
#define WAVES_PER_BLOCK 8

// ---------------------------------------------------------------------------
// Fused dual GEMM: out_self = x @ Wself^T + bself ; y_neigh = x @ Wneigh^T
// One wave computes a 16-row block of both outputs using V_WMMA_F32_16X16X4_F32.
// A-frag layout (16x4 f32): lanes 0-15 -> M=0..15 / K={0,1}; lanes 16-31 -> K={2,3}
// B-frag layout (4x16):     lanes 0-15 -> N=0..15 / K={0,1}; lanes 16-31 -> K={2,3}
//   B[k][n] = W[n][k]  (W row-major) -> per-lane float2 is contiguous in W's row.
// C/D layout: VGPR v, lane half h -> M = v + 8h, N = lane%16.
// ---------------------------------------------------------------------------
__global__ __launch_bounds__(256) void gnn_dual_gemm_wmma(
    const float* __restrict__ x,
    const float* __restrict__ Wself, const float* __restrict__ bself,
    const float* __restrict__ Wneigh,
    float* __restrict__ out_self, float* __restrict__ y_neigh, int n)
{
    const int lane    = threadIdx.x & 31;
    const int waveId  = (blockIdx.x * WAVES_PER_BLOCK) + (threadIdx.x >> 5);
    const int row0    = waveId * 16;
    if (row0 >= n) return;                 // wave-uniform: EXEC stays all-1s

    const int half = lane >> 4;            // 0 or 1
    const int lm   = lane & 15;            // 0..15

    // Cache the 16x128 A tile: 32 K-steps, one float2 per lane per step.
    v2f afrag[32];
    const float* xrow = x + (size_t)(row0 + lm) * DD + half * 2;
    #pragma unroll
    for (int s = 0; s < 32; ++s)
        afrag[s] = *(const v2f*)(xrow + s * 4);

    #pragma unroll 1
    for (int j = 0; j < 8; ++j) {          // 8 column tiles of 16
        v8f acc_s = {};
        v8f acc_n = {};
        const float* wsrow = Wself  + (size_t)(j * 16 + lm) * DD + half * 2;
        const float* wnrow = Wneigh + (size_t)(j * 16 + lm) * DD + half * 2;
        #pragma unroll
        for (int s = 0; s < 32; ++s) {     // K = 128 in steps of 4
            v2f bs = *(const v2f*)(wsrow + s * 4);
            v2f bn = *(const v2f*)(wnrow + s * 4);
            acc_s = __builtin_amdgcn_wmma_f32_16x16x4_f32(
                false, afrag[s], false, bs, (short)0, acc_s, false, false);
            acc_n = __builtin_amdgcn_wmma_f32_16x16x4_f32(
                false, afrag[s], false, bn, (short)0, acc_n, false, false);
        }
        const float bias = bself[j * 16 + lm];
        #pragma unroll
        for (int v = 0; v < 8; ++v) {
            const int m = v + half * 8;
            const size_t idx = (size_t)(row0 + m) * DD + j * 16 + lm;
            out_self[idx] = acc_s[v] + bias;
            y_neigh[idx]  = acc_n[v];
        }
    }
}

// ---------------------------------------------------------------------------
// Zero the agg + count region of workspace (contiguous N*(128+1) floats).
// ---------------------------------------------------------------------------
__global__ void gnn_zero(float* __restrict__ p, int total)
{
    int i = blockIdx.x * blockDim.x + threadIdx.x;
    if (i < total) p[i] = 0.0f;
}

// ---------------------------------------------------------------------------
// Edge scatter: one wave per edge. 32 lanes x float4 = 128-wide row.
// agg[row] += y[col]; cnt[row] += 1
// ---------------------------------------------------------------------------
__global__ __launch_bounds__(256) void gnn_scatter(
    const long long* __restrict__ ei,   // [2, E] int64: row = ei[e], col = ei[E+e]
    const float* __restrict__ y,
    float* __restrict__ agg, float* __restrict__ cnt, int e_cnt)
{
    const int wave = (int)((blockIdx.x * blockDim.x + threadIdx.x) >> 5);
    const int lane = threadIdx.x & 31;
    if (wave >= e_cnt) return;

    const long long row = ei[wave];
    const long long col = ei[(size_t)e_cnt + wave];

    const float4 v = *(const float4*)(y + (size_t)col * DD + lane * 4);
    float* dst = agg + (size_t)row * DD + lane * 4;
    atomicAdd(dst + 0, v.x);
    atomicAdd(dst + 1, v.y);
    atomicAdd(dst + 2, v.z);
    atomicAdd(dst + 3, v.w);
    if (lane == 0) atomicAdd(cnt + row, 1.0f);
}

// ---------------------------------------------------------------------------
// Finalize: out += agg / cnt + b_neigh   (exactly 0 contribution when cnt==0)
// ---------------------------------------------------------------------------
__global__ void gnn_finalize(
    const float* __restrict__ agg, const float* __restrict__ cnt,
    const float* __restrict__ bneigh, float* __restrict__ out, int total)
{
    int i = blockIdx.x * blockDim.x + threadIdx.x;
    if (i >= total) return;
    const int node = i >> 7;     // / 128
    const int c    = i & 127;
    const float dcount = cnt[node];
    const float add = (dcount > 0.0f) ? (agg[i] / dcount + bneigh[c]) : 0.0f;
    out[i] += add;
}

extern "C" void kernel_launch(void* const* d_in, const int* in_sizes, int n_in,
                              void* d_out, int out_size, void* d_ws, size_t ws_size,
                              hipStream_t stream)
{
    const float*     x      = (const float*)d_in[0];
    const long long* ei     = (const long long*)d_in[1];   // int64 edge index
    const float*     Wself  = (const float*)d_in[2];
    const float*     bself  = (const float*)d_in[3];
    const float*     Wneigh = (const float*)d_in[4];
    const float*     bneigh = (const float*)d_in[5];
    float*           out    = (float*)d_out;

    const int n = in_sizes[0] / DD;          // 10000
    const int e = in_sizes[1] / 2;           // 640000

    // Workspace layout: y [n*128] | agg [n*128] | cnt [n]
    float* y   = (float*)d_ws;
    float* agg = y + (size_t)n * DD;
    float* cnt = agg + (size_t)n * DD;

    // 1) Fused dual WMMA GEMM (16 rows per wave, 8 waves per block)
    {
        const int rowBlocks = (n + 15) / 16;
        const int blocks = (rowBlocks + WAVES_PER_BLOCK - 1) / WAVES_PER_BLOCK;
        gnn_dual_gemm_wmma<<<blocks, 256, 0, stream>>>(x, Wself, bself, Wneigh,
                                                       out, y, n);
    }

    // 2) Zero agg + cnt (contiguous region of n*(128+1) floats)
    {
        const int total = n * (DD + 1);
        gnn_zero<<<(total + 255) / 256, 256, 0, stream>>>(agg, total);
    }

    // 3) Edge scatter with fp32 global atomics (one wave per edge)
    {
        const long long threads = (long long)e * 32;
        const int blocks = (int)((threads + 255) / 256);
        gnn_scatter<<<blocks, 256, 0, stream>>>(ei, y, agg, cnt, e);
    }

    // 4) Finalize: mean-aggregate + neighbor bias
    {
        const int total = n * DD;
        gnn_finalize<<<(total + 255) / 256, 256, 0, stream>>>(agg, cnt, bneigh,
                                                              out, total);
    }
}